// QuantumKernelMethod_65481071410383
// MI455X (gfx1250) — compile-verified
//
#include <hip/hip_runtime.h>
#include <hip/hip_bf16.h>

// QuantumKernelMethod — analytic collapse:
//   out[b] = | prod_q cos( (x[b][q] - y[b][q]) * 0.5 ) |
// (CNOT chain and per-qubit SU(2) rotations are unitary and cancel in the
//  overlap; params are mathematically irrelevant to the output.)
//
// Pure streaming kernel: 32 B in + 4 B out per element, HBM-bound
// (~36 MB total -> ~1.55 us at 23.3 TB/s). Uses b128 coalesced loads with
// non-temporal hints (single-pass data, keep L2/WGP$ clean) and the hardware
// v_cos_f32 transcendental.

typedef float v4f __attribute__((ext_vector_type(4)));

__global__ __launch_bounds__(256) void QuantumKernelMethod_overlap_kernel(
    const v4f* __restrict__ x4,
    const v4f* __restrict__ y4,
    float* __restrict__ out,
    int B)
{
    int i = blockIdx.x * blockDim.x + threadIdx.x;
    if (i >= B) return;

    // One 128-bit NT load per input vector (4 qubit angles each).
    v4f xv = __builtin_nontemporal_load(x4 + i);
    v4f yv = __builtin_nontemporal_load(y4 + i);

    float p;
    p  = __cosf(0.5f * (xv.x - yv.x));
    p *= __cosf(0.5f * (xv.y - yv.y));
    p *= __cosf(0.5f * (xv.z - yv.z));
    p *= __cosf(0.5f * (xv.w - yv.w));

    __builtin_nontemporal_store(fabsf(p), out + i);
}

extern "C" void kernel_launch(void* const* d_in, const int* in_sizes, int n_in,
                              void* d_out, int out_size, void* d_ws, size_t ws_size,
                              hipStream_t stream)
{
    (void)n_in; (void)d_ws; (void)ws_size;

    const v4f* x4 = (const v4f*)d_in[0];   // x: (B, 4) float32
    const v4f* y4 = (const v4f*)d_in[1];   // y: (B, 4) float32
    // d_in[2] = params (4,3): mathematically cancels out of |<psi_x|psi_y>|.

    float* out = (float*)d_out;            // (B,) float32
    const int B = in_sizes[0] / 4;         // 1048576 for the reference shapes

    const int block = 256;                 // 8 wave32's per workgroup
    const int grid  = (B + block - 1) / block;
    QuantumKernelMethod_overlap_kernel<<<grid, block, 0, stream>>>(x4, y4, out, B);
}